// MultiHeadAttention_28286654611885
// MI455X (gfx1250) — compile-verified
//
#include <hip/hip_runtime.h>
#include <hip/hip_bf16.h>
#include <math.h>

// Problem constants (fixed by the reference)
#define D_MODEL 1024
#define N_HEADS 16
#define D_K     64
#define BATCH   2
#define SEQ     2048
#define M_TOTAL (BATCH * SEQ)   // 4096 rows

typedef __attribute__((ext_vector_type(2))) float v2f;
typedef __attribute__((ext_vector_type(8))) float v8f;

// D(16x16,f32) = A(16x4,f32) * B(4x16,f32) + C
// 8-arg form: (neg_a, A, neg_b, B, c_mod, C, reuse_a, reuse_b)
__device__ __forceinline__ v8f wmma_f32(v2f a, v2f b, v8f c) {
    return __builtin_amdgcn_wmma_f32_16x16x4_f32(false, a, false, b,
                                                 (short)0, c, false, false);
}

// ---------------------------------------------------------------------------
// GEMM: C[M,N] = A[M,K] @ W[K,N] + bias (+ optional residual)
// Block = 256 threads (8 waves), tile 128x64, BK=32.
// Wave w computes a 32x32 quadrant: rows (w&3)*32..+31, cols (w>>2)*32..+31,
// as a 2x2 grid of 16x16 WMMA accumulators (4 frag loads -> 4 WMMAs / chunk).
// W tile is stored TRANSPOSED in LDS so each B-frag is a contiguous dword
// pair -> single ds_load_2addr_b32 offset1:1, no repack movs.
// ---------------------------------------------------------------------------
#define BM 128
#define BN 64
#define BK 32

__global__ __launch_bounds__(256)
void gemm_bias_kernel(const float* __restrict__ A, const float* __restrict__ W,
                      const float* __restrict__ bias,
                      const float* __restrict__ residual,
                      float* __restrict__ C, int M, int N, int K)
{
    __shared__ float As[BM][BK + 1];   // 128 x 33  (row-major: A-frags contiguous)
    __shared__ float Wt[BN][BK + 1];   // 64 x 33   (transposed: B-frags contiguous)

    const int tid  = threadIdx.x;
    const int wave = tid >> 5;
    const int lane = tid & 31;
    const int l16  = lane & 15;
    const int hi   = lane >> 4;        // which 16-lane half

    const int rowBase = blockIdx.y * BM;
    const int colBase = blockIdx.x * BN;
    const int sr = (wave & 3) * 32;    // wave's row quadrant
    const int sc = (wave >> 2) * 32;   // wave's col quadrant

    v8f acc00 = {}, acc01 = {}, acc10 = {}, acc11 = {};

    for (int k0 = 0; k0 < K; k0 += BK) {
        __syncthreads();  // previous compute done before LDS overwrite
        // Stage A tile: 128x32 floats = 1024 float4, 4 per thread
        #pragma unroll
        for (int t = tid; t < (BM * BK / 4); t += 256) {
            const int row = t >> 3;            // / (BK/4)
            const int c4  = (t & 7) * 4;
            const float4 av = *(const float4*)(A + (size_t)(rowBase + row) * K + k0 + c4);
            As[row][c4 + 0] = av.x; As[row][c4 + 1] = av.y;
            As[row][c4 + 2] = av.z; As[row][c4 + 3] = av.w;
            if (k0 + BK < K)
                __builtin_prefetch(A + (size_t)(rowBase + row) * K + k0 + BK + c4, 0, 1);
        }
        // Stage W tile transposed: W[k0+row][colBase+c] -> Wt[c][row]
        #pragma unroll
        for (int t = tid; t < (BK * BN / 4); t += 256) {
            const int row = t >> 4;            // / (BN/4)
            const int c4  = (t & 15) * 4;
            const float4 wv = *(const float4*)(W + (size_t)(k0 + row) * N + colBase + c4);
            Wt[c4 + 0][row] = wv.x; Wt[c4 + 1][row] = wv.y;
            Wt[c4 + 2][row] = wv.z; Wt[c4 + 3][row] = wv.w;
            if (k0 + BK < K)
                __builtin_prefetch(W + (size_t)(k0 + BK + row) * N + colBase + c4, 0, 1);
        }
        __syncthreads();

        // A-frag: A[m][k], m = l16, k = kk + 2*hi + r  (r = vgpr 0/1)
        // B-frag: B[k][n] = Wt[n][k], n = l16, k = kk + 2*hi + r
        #pragma unroll
        for (int kk = 0; kk < BK; kk += 4) {
            const int ka = kk + 2 * hi;
            v2f a0; a0.x = As[sr +      l16][ka]; a0.y = As[sr +      l16][ka + 1];
            v2f a1; a1.x = As[sr + 16 + l16][ka]; a1.y = As[sr + 16 + l16][ka + 1];
            v2f b0; b0.x = Wt[sc +      l16][ka]; b0.y = Wt[sc +      l16][ka + 1];
            v2f b1; b1.x = Wt[sc + 16 + l16][ka]; b1.y = Wt[sc + 16 + l16][ka + 1];
            acc00 = wmma_f32(a0, b0, acc00);
            acc01 = wmma_f32(a0, b1, acc01);
            acc10 = wmma_f32(a1, b0, acc10);
            acc11 = wmma_f32(a1, b1, acc11);
        }
    }

    // C/D layout: vgpr r, half hi -> row = 8*hi + r ; col = l16
    #pragma unroll
    for (int r = 0; r < 8; ++r) {
        const int r0 = rowBase + sr + 8 * hi + r;        // acc00/acc01 rows
        const int r1 = r0 + 16;                          // acc10/acc11 rows
        const int c0 = colBase + sc + l16;
        const int c1 = c0 + 16;
        float o00 = acc00[r] + bias[c0];
        float o01 = acc01[r] + bias[c1];
        float o10 = acc10[r] + bias[c0];
        float o11 = acc11[r] + bias[c1];
        if (residual != nullptr) {
            o00 += residual[(size_t)r0 * N + c0];
            o01 += residual[(size_t)r0 * N + c1];
            o10 += residual[(size_t)r1 * N + c0];
            o11 += residual[(size_t)r1 * N + c1];
        }
        C[(size_t)r0 * N + c0] = o00;
        C[(size_t)r0 * N + c1] = o01;
        C[(size_t)r1 * N + c0] = o10;
        C[(size_t)r1 * N + c1] = o11;
    }
}

// ---------------------------------------------------------------------------
// Flash attention: per (batch, head), streaming over keys in blocks of 16.
// Block = 128 threads (4 waves); each wave owns 16 query rows.
// Q/K/V/Ctx all stored [B, S, D_MODEL] with head h at columns h*64..h*64+63.
// V tile stored TRANSPOSED in LDS so the P@V B-frags are contiguous pairs.
// ---------------------------------------------------------------------------
#define QB 64
#define KB 16

__global__ __launch_bounds__(128)
void attention_kernel(const float* __restrict__ Q, const float* __restrict__ K,
                      const float* __restrict__ V, float* __restrict__ Ctx)
{
    __shared__ float Ks[KB][D_K + 1];   // row-major: S=QK^T B-frags contiguous
    __shared__ float Vt[D_K][KB + 1];   // transposed: P@V B-frags contiguous
    __shared__ float Ps[4][16][17];     // per-wave P staging (C-layout -> A-layout)

    const int tid  = threadIdx.x;
    const int wave = tid >> 5;
    const int lane = tid & 31;
    const int l16  = lane & 15;
    const int hi   = lane >> 4;

    const int b = blockIdx.z;
    const int h = blockIdx.y;
    const int qBase = blockIdx.x * QB + wave * 16;           // query row in sequence
    const size_t headOff = (size_t)b * SEQ * D_MODEL + (size_t)h * D_K;

    // Load this wave's Q strip (16 x 64) into A-frag layout, pre-scaled by 1/sqrt(Dk).
    // qreg[2c+r] = Q[qBase + l16][4c + 2*hi + r] * 0.125
    float qreg[32];
    {
        const float* qrow = Q + headOff + (size_t)(qBase + l16) * D_MODEL;
        #pragma unroll
        for (int c = 0; c < 16; ++c) {
            const int k = 4 * c + 2 * hi;
            qreg[2 * c + 0] = qrow[k + 0] * 0.125f;
            qreg[2 * c + 1] = qrow[k + 1] * 0.125f;
        }
    }

    v8f o0 = {}, o1 = {}, o2 = {}, o3 = {};
    float mrow[8], lrow[8];
    #pragma unroll
    for (int r = 0; r < 8; ++r) { mrow[r] = -1e30f; lrow[r] = 0.0f; }

    for (int j = 0; j < SEQ; j += KB) {
        __syncthreads();  // previous iteration done with Ks/Vt
        // Cooperative stage of K (row-major) and V (transposed) tiles
        #pragma unroll
        for (int t = tid; t < (KB * D_K / 4); t += 128) {
            const int row = t >> 4;
            const int c4  = (t & 15) * 4;
            const float4 kv = *(const float4*)(K + headOff + (size_t)(j + row) * D_MODEL + c4);
            Ks[row][c4 + 0] = kv.x; Ks[row][c4 + 1] = kv.y;
            Ks[row][c4 + 2] = kv.z; Ks[row][c4 + 3] = kv.w;
            const float4 vv = *(const float4*)(V + headOff + (size_t)(j + row) * D_MODEL + c4);
            Vt[c4 + 0][row] = vv.x; Vt[c4 + 1][row] = vv.y;
            Vt[c4 + 2][row] = vv.z; Vt[c4 + 3][row] = vv.w;
        }
        __syncthreads();

        // S = Q @ K^T : B[k][n] = K[j+n][k] -> b-frag reads Ks[l16][4c+2hi (+1)]
        v8f s = {};
        #pragma unroll
        for (int c = 0; c < 16; ++c) {
            const int ka = 4 * c + 2 * hi;
            v2f a;  a.x = qreg[2 * c]; a.y = qreg[2 * c + 1];
            v2f bf; bf.x = Ks[l16][ka]; bf.y = Ks[l16][ka + 1];
            s = wmma_f32(a, bf, s);
        }

        // Online softmax per row (row m = 8*hi + r lives in 16 lanes of one half)
        #pragma unroll
        for (int r = 0; r < 8; ++r) {
            float sv = s[r];
            float mx = sv;
            mx = fmaxf(mx, __shfl_xor(mx, 1, 16));
            mx = fmaxf(mx, __shfl_xor(mx, 2, 16));
            mx = fmaxf(mx, __shfl_xor(mx, 4, 16));
            mx = fmaxf(mx, __shfl_xor(mx, 8, 16));
            const float mnew = fmaxf(mrow[r], mx);
            const float corr = __expf(mrow[r] - mnew);
            const float p    = __expf(sv - mnew);
            float ps = p;
            ps += __shfl_xor(ps, 1, 16);
            ps += __shfl_xor(ps, 2, 16);
            ps += __shfl_xor(ps, 4, 16);
            ps += __shfl_xor(ps, 8, 16);
            lrow[r] = lrow[r] * corr + ps;
            mrow[r] = mnew;
            o0[r] *= corr; o1[r] *= corr; o2[r] *= corr; o3[r] *= corr;
            Ps[wave][8 * hi + r][l16] = p;   // C-layout -> LDS
        }
        __syncthreads();  // Ps visible before A-frag reads

        // O += P @ V : A[m][k]=P[m][k] (m=l16), B[k][n]=V[j+k][16q+n]=Vt[16q+n][k]
        #pragma unroll
        for (int c = 0; c < 4; ++c) {
            const int ka = 4 * c + 2 * hi;
            v2f a; a.x = Ps[wave][l16][ka]; a.y = Ps[wave][l16][ka + 1];
            v2f b0, b1, b2, b3;
            b0.x = Vt[ 0 + l16][ka]; b0.y = Vt[ 0 + l16][ka + 1];
            b1.x = Vt[16 + l16][ka]; b1.y = Vt[16 + l16][ka + 1];
            b2.x = Vt[32 + l16][ka]; b2.y = Vt[32 + l16][ka + 1];
            b3.x = Vt[48 + l16][ka]; b3.y = Vt[48 + l16][ka + 1];
            o0 = wmma_f32(a, b0, o0);
            o1 = wmma_f32(a, b1, o1);
            o2 = wmma_f32(a, b2, o2);
            o3 = wmma_f32(a, b3, o3);
        }
    }

    // Normalize and write context (heads re-concatenated into [B,S,D])
    #pragma unroll
    for (int r = 0; r < 8; ++r) {
        const int m = 8 * hi + r;
        const float inv = 1.0f / lrow[r];
        float* dst = Ctx + headOff + (size_t)(qBase + m) * D_MODEL;
        dst[ 0 + l16] = o0[r] * inv;
        dst[16 + l16] = o1[r] * inv;
        dst[32 + l16] = o2[r] * inv;
        dst[48 + l16] = o3[r] * inv;
    }
}

// ---------------------------------------------------------------------------
// In-place LayerNorm over rows of length 1024. One block (256 thr) per row;
// every thread reads its 4 elements before any write -> in-place is safe.
// ---------------------------------------------------------------------------
__global__ __launch_bounds__(256)
void layernorm_kernel(float* __restrict__ Y, const float* __restrict__ gamma,
                      const float* __restrict__ beta)
{
    __shared__ float sbuf[8];
    const int tid  = threadIdx.x;
    const int wave = tid >> 5;
    const int lane = tid & 31;
    float* y = Y + (size_t)blockIdx.x * D_MODEL;

    float v[4];
    float s = 0.0f;
    #pragma unroll
    for (int i = 0; i < 4; ++i) { v[i] = y[tid + i * 256]; s += v[i]; }
    for (int m = 16; m >= 1; m >>= 1) s += __shfl_xor(s, m, 32);
    if (lane == 0) sbuf[wave] = s;
    __syncthreads();
    float tot = 0.0f;
    #pragma unroll
    for (int i = 0; i < 8; ++i) tot += sbuf[i];
    const float mean = tot * (1.0f / D_MODEL);
    __syncthreads();

    float vs = 0.0f;
    #pragma unroll
    for (int i = 0; i < 4; ++i) { const float d = v[i] - mean; vs += d * d; }
    for (int m = 16; m >= 1; m >>= 1) vs += __shfl_xor(vs, m, 32);
    if (lane == 0) sbuf[wave] = vs;
    __syncthreads();
    float tot2 = 0.0f;
    #pragma unroll
    for (int i = 0; i < 8; ++i) tot2 += sbuf[i];
    const float inv = rsqrtf(tot2 * (1.0f / D_MODEL) + 1e-5f);

    #pragma unroll
    for (int i = 0; i < 4; ++i) {
        const int col = tid + i * 256;
        y[col] = (v[i] - mean) * inv * gamma[col] + beta[col];
    }
}

// ---------------------------------------------------------------------------
extern "C" void kernel_launch(void* const* d_in, const int* in_sizes, int n_in,
                              void* d_out, int out_size, void* d_ws, size_t ws_size,
                              hipStream_t stream) {
    (void)in_sizes; (void)n_in; (void)out_size; (void)ws_size;
    const float* x     = (const float*)d_in[0];
    const float* Wq    = (const float*)d_in[1];
    const float* bq    = (const float*)d_in[2];
    const float* Wk    = (const float*)d_in[3];
    const float* bk    = (const float*)d_in[4];
    const float* Wv    = (const float*)d_in[5];
    const float* bv    = (const float*)d_in[6];
    const float* Wo    = (const float*)d_in[7];
    const float* bo    = (const float*)d_in[8];
    const float* gamma = (const float*)d_in[9];
    const float* beta  = (const float*)d_in[10];
    float* out = (float*)d_out;

    const size_t elems = (size_t)M_TOTAL * D_MODEL;   // 4M floats / buffer
    float* Q   = (float*)d_ws;
    float* Kb  = Q  + elems;
    float* Vb  = Kb + elems;
    float* Ctx = Vb + elems;

    const dim3 gblk(256);
    const dim3 ggrid(D_MODEL / BN, M_TOTAL / BM);     // (16, 32)

    gemm_bias_kernel<<<ggrid, gblk, 0, stream>>>(x, Wq, bq, nullptr, Q,  M_TOTAL, D_MODEL, D_MODEL);
    gemm_bias_kernel<<<ggrid, gblk, 0, stream>>>(x, Wk, bk, nullptr, Kb, M_TOTAL, D_MODEL, D_MODEL);
    gemm_bias_kernel<<<ggrid, gblk, 0, stream>>>(x, Wv, bv, nullptr, Vb, M_TOTAL, D_MODEL, D_MODEL);

    attention_kernel<<<dim3(SEQ / QB, N_HEADS, BATCH), 128, 0, stream>>>(Q, Kb, Vb, Ctx);

    // y = x + Ctx @ Wo + bo  (written directly to d_out), then LN in place
    gemm_bias_kernel<<<ggrid, gblk, 0, stream>>>(Ctx, Wo, bo, x, out, M_TOTAL, D_MODEL, D_MODEL);
    layernorm_kernel<<<M_TOTAL, 256, 0, stream>>>(out, gamma, beta);
}